// FlaxParallelExperts_42880953483995
// MI455X (gfx1250) — compile-verified
//
#include <hip/hip_runtime.h>

typedef __attribute__((ext_vector_type(2))) float v2f;
typedef __attribute__((ext_vector_type(8))) float v8f;

// Problem constants (from reference)
constexpr int kE = 8;     // experts
constexpr int kC = 8;     // chunks
constexpr int kM = 2048;  // INPUT_SIZE (i)  -> GEMM M
constexpr int kK = 2048;  // OUTPUT_SIZE (o) -> GEMM K (contraction)
constexpr int kN = 1024;  // N_COLS (n)      -> GEMM N

// Tiling
constexpr int kBM = 128;
constexpr int kBN = 128;
constexpr int kBK = 32;
constexpr int kLD = kBK + 4;      // 36 floats -> 144B rows: 16B-aligned, conflict-free frag reads
constexpr int kIters = kK / kBK;  // 64
constexpr int kAsyncPerTile = 20; // 4x b128 (A) + 16x b32 (B) per thread

// ---- CDNA5 async global->LDS copies, GVS addressing (SGPR base + 32-bit VGPR offset) ----
__device__ __forceinline__ void async_b128(unsigned lds_off, unsigned voff, const float* base) {
    asm volatile("global_load_async_to_lds_b128 %0, %1, %2"
                 :: "v"(lds_off), "v"(voff), "s"((unsigned long long)(uintptr_t)base)
                 : "memory");
}
__device__ __forceinline__ void async_b32(unsigned lds_off, unsigned voff, const float* base) {
    asm volatile("global_load_async_to_lds_b32 %0, %1, %2"
                 :: "v"(lds_off), "v"(voff), "s"((unsigned long long)(uintptr_t)base)
                 : "memory");
}

__global__ __launch_bounds__(256)
void FlaxParallelExperts_cross_gemm(const float* __restrict__ X,  // [C][K][N] inputs
                                    const float* __restrict__ W,  // [E][M][K] weight
                                    float* __restrict__ Out)      // [C*E][M][N]
{
    // Double-buffered LDS tiles. A row-major [m][k]; B transposed [n][k].
    __shared__ float sA[2][kBM * kLD];   // 2 x 18432 B
    __shared__ float sB[2][kBN * kLD];   // 2 x 18432 B

    const int tid  = threadIdx.x;
    const int lane = tid & 31;
    const int wave = tid >> 5;
    const int wm   = wave >> 1;   // 0..3 (m direction)
    const int wn   = wave & 1;    // 0..1 (n direction)
    const int hl   = lane >> 4;   // K-pair select in WMMA f32 16x16x4 layout
    const int ml   = lane & 15;   // row/col within 16

    const int n0    = blockIdx.x * kBN;
    const int i0    = blockIdx.y * kBM;
    const int batch = blockIdx.z;      // c*8 + e
    const int c     = batch >> 3;
    const int e     = batch & 7;

    const float* Ag = W + (size_t)e * kM * kK + (size_t)i0 * kK;  // 128 rows, stride kK
    const float* Bg = X + (size_t)c * kK * kN + n0;               // kK rows, stride kN
    float*       Dg = Out + (size_t)batch * kM * kN + (size_t)i0 * kN + n0;

    v8f acc[2][4];
    {
        v8f z = {};
        #pragma unroll
        for (int a = 0; a < 2; ++a)
            #pragma unroll
            for (int b = 0; b < 4; ++b)
                acc[a][b] = z;
    }

    // Issue one K-tile. GVS: per-thread byte offsets are loop-invariant VGPRs,
    // the k0-dependent base is a wave-uniform SGPR pair (scalar add per tile).
    auto issue_tile = [&](int buf, int k0) {
        const float* aBase = Ag + k0;                  // column step within A rows
        const float* bBase = Bg + (size_t)k0 * kN;     // row step within B
        #pragma unroll
        for (int u = 0; u < 4; ++u) {
            const int f  = tid + u * 256;              // 0..1023 float4 chunks of A (128x32)
            const int ar = f >> 3;                     // 0..127
            const int ac = (f & 7) << 2;               // 0,4,...,28
            async_b128((unsigned)(uintptr_t)&sA[buf][ar * kLD + ac],
                       (unsigned)((ar * kK + ac) * 4), aBase);
        }
        #pragma unroll
        for (int u = 0; u < 16; ++u) {
            const int f  = tid + u * 256;              // 0..4095 elements of B (32x128)
            const int bk = f >> 7;                     // 0..31
            const int bn = f & 127;                    // consecutive across lanes -> coalesced
            async_b32((unsigned)(uintptr_t)&sB[buf][bn * kLD + bk],
                      (unsigned)((bk * kN + bn) * 4), bBase);
        }
    };

    issue_tile(0, 0);

    for (int it = 0; it < kIters; ++it) {
        const int cur = it & 1;

        if (it + 1 < kIters) {
            issue_tile(cur ^ 1, (it + 1) * kBK);
            // 20 ops of the next tile in flight; ASYNCcnt completes in order =>
            // waiting for <=20 outstanding guarantees the current tile landed.
            asm volatile("s_wait_asynccnt 20" ::: "memory");
        } else {
            asm volatile("s_wait_asynccnt 0" ::: "memory");
        }
        __syncthreads();

        #pragma unroll
        for (int kk = 0; kk < kBK; kk += 4) {
            v2f afrag[2];
            #pragma unroll
            for (int tm = 0; tm < 2; ++tm) {
                const int mrow = wm * 32 + tm * 16 + ml;
                // A lane layout (16x4 f32): M = lane%16, K = 2*(lane/16)+j (contiguous v2f)
                afrag[tm] = *(const v2f*)&sA[cur][mrow * kLD + kk + hl * 2];
            }
            v2f bfrag[4];
            #pragma unroll
            for (int tn = 0; tn < 4; ++tn) {
                const int ncol = wn * 64 + tn * 16 + ml;
                // B lane layout (4x16 f32): N = lane%16, K = 2*(lane/16)+j;
                // sB is [n][k] so this is a contiguous, pair-aligned v2f -> ds_load_b64
                bfrag[tn] = *(const v2f*)&sB[cur][ncol * kLD + kk + hl * 2];
            }
            #pragma unroll
            for (int tm = 0; tm < 2; ++tm)
                #pragma unroll
                for (int tn = 0; tn < 4; ++tn)
                    acc[tm][tn] = __builtin_amdgcn_wmma_f32_16x16x4_f32(
                        /*neg_a=*/false, afrag[tm],
                        /*neg_b=*/false, bfrag[tn],
                        /*c_mod=*/(short)0, acc[tm][tn],
                        /*reuse_a=*/false, /*reuse_b=*/false);
        }
        __syncthreads();
    }

    // Epilogue: D VGPR r -> row (tile + 8*(lane/16) + r), col (lane%16).
    // Output is a 512MB single-use stream: non-temporal stores keep A/B hot in L2.
    #pragma unroll
    for (int tm = 0; tm < 2; ++tm) {
        #pragma unroll
        for (int tn = 0; tn < 4; ++tn) {
            const int ncol = wn * 64 + tn * 16 + ml;
            #pragma unroll
            for (int r = 0; r < 8; ++r) {
                const int mrow = wm * 32 + tm * 16 + hl * 8 + r;
                __builtin_nontemporal_store(acc[tm][tn][r], Dg + (size_t)mrow * kN + ncol);
            }
        }
    }
}

extern "C" void kernel_launch(void* const* d_in, const int* in_sizes, int n_in,
                              void* d_out, int out_size, void* d_ws, size_t ws_size,
                              hipStream_t stream) {
    (void)in_sizes; (void)n_in; (void)out_size; (void)d_ws; (void)ws_size;
    const float* X = (const float*)d_in[0];   // inputs  [8][2048][1024] fp32
    const float* W = (const float*)d_in[1];   // weight  [8][2048][2048] fp32
    float* Out = (float*)d_out;               // [64][2048][1024] fp32

    dim3 grid(kN / kBN, kM / kBM, kC * kE);   // (8, 16, 64)
    dim3 block(256);
    hipLaunchKernelGGL(FlaxParallelExperts_cross_gemm, grid, block, 0, stream, X, W, Out);
}